// MyDistance_22497038696716
// MI455X (gfx1250) — compile-verified
//
#include <hip/hip_runtime.h>

// ---------------------------------------------------------------------------
// radius_graph (per-molecule KNN within cutoff) for MI455X / gfx1250
//   N_MOL=256, MOL_SIZE=512, MAX_NBR=32, cutoff=5.0 (d2 < 25), loop=false
// Output layout in d_out (concatenated, reference return order):
//   int32  edge_index[2][E]   (row = neighbor, col = center, pad = -1)
//   float  edge_weight[E]
//   float  edge_vec[E][3]
//   uint8  mask[E]            (bool)
// with E = N * MAX_NBR = 131072*32 = 4194304
//
// Per-tile distances via V_WMMA_F32_16X16X4_F32:
//   A row m = (-2x_m, -2y_m, -2z_m, sq_m), B col n = (x_n, y_n, z_n, 1)
//   => D[m][n] = sq_m - 2 p_m.p_n ; add sq_n (lane scalar) -> d2.
// ---------------------------------------------------------------------------

#define NMOL   256
#define MSIZE  512
#define KNBR   32
#define CUT2   25.0f

typedef __attribute__((ext_vector_type(2))) float v2f;
typedef __attribute__((ext_vector_type(8))) float v8f;

__global__ __launch_bounds__(1024, 1)
void radius_knn_kernel(const float* __restrict__ pos,
                       int* __restrict__ ei_row,
                       int* __restrict__ ei_col,
                       float* __restrict__ ew,
                       float* __restrict__ evec,
                       unsigned char* __restrict__ emask)
{
    __shared__ float4 sP[MSIZE];   // (x, y, z, |p|^2)
    __shared__ float4 sA[MSIZE];   // (-2x, -2y, -2z, |p|^2)  -> WMMA A operand

    const int mol = blockIdx.x;
    const int tid = threadIdx.x;

    // ---- stage positions for this molecule into LDS --------------------
    if (tid < MSIZE) {
        const float* p = pos + ((size_t)mol * MSIZE + tid) * 3;
        float x = p[0], y = p[1], z = p[2];
        float sq = x * x + y * y + z * z;
        sP[tid] = make_float4(x, y, z, sq);
        sA[tid] = make_float4(-2.0f * x, -2.0f * y, -2.0f * z, sq);
    }
    __syncthreads();

    const int lane = tid & 31;
    const int wave = tid >> 5;       // 0..31, wave w owns centers [16w,16w+16)
    const int half = lane >> 4;      // 0: rows 0-7 of each tile, 1: rows 8-15
    const int ln   = lane & 15;
    const int cl   = (wave << 4) + ln;   // this lane's center (local id)

    const float4 pc  = sP[cl];
    const float  sqc = pc.w;

    // B operand (loop invariant): column n = center, col = (x,y,z,1)
    // 32-bit B layout: lanes 0-15 hold K=0..1, lanes 16-31 hold K=2..3.
    v2f bv;
    if (half == 0) { bv.x = pc.x; bv.y = pc.y; }
    else           { bv.x = pc.z; bv.y = 1.0f; }

    // private sorted top-32 list over this lane's candidate stream
    float dist[KNBR];
    int   nidx[KNBR];
#pragma unroll
    for (int i = 0; i < KNBR; ++i) { dist[i] = __builtin_inff(); nidx[i] = -1; }

    // ---- sweep neighbor tiles; software-pipeline the A-operand ds_load --
    v2f av = ((const v2f*)(&sA[ln]))[half];          // tile 0
    for (int tb = 0; tb < MSIZE; tb += 16) {
        v8f c = __builtin_amdgcn_wmma_f32_16x16x4_f32(
                    /*neg_a=*/false, av, /*neg_b=*/false, bv,
                    /*c_mod=*/(short)0,
                    v8f{0.f,0.f,0.f,0.f,0.f,0.f,0.f,0.f},
                    /*reuse_a=*/false, /*reuse_b=*/false);

        // prefetch next tile's A while candidates are processed
        if (tb + 16 < MSIZE)
            av = ((const v2f*)(&sA[tb + 16 + ln]))[half];

#pragma unroll
        for (int k = 0; k < 8; ++k) {
            const int   m  = tb + half * 8 + k;     // candidate neighbor
            const float d2 = c[k] + sqc;
            if (d2 < CUT2 && m != cl && d2 < dist[KNBR - 1]) {
                // unrolled sorted insertion (static indices -> stays in VGPRs)
#pragma unroll
                for (int j = KNBR - 1; j >= 1; --j) {
                    bool lt    = d2 < dist[j];
                    bool shift = d2 < dist[j - 1];
                    float nd = shift ? dist[j - 1] : d2;
                    int   ni = shift ? nidx[j - 1] : m;
                    dist[j] = lt ? nd : dist[j];
                    nidx[j] = lt ? ni : nidx[j];
                }
                if (d2 < dist[0]) { dist[0] = d2; nidx[0] = m; }
            }
        }
    }

    // ---- merge lane pair: keep-lowest-32 of two sorted 32-lists ---------
    // md[i] = min(A[i], B[31-i]) (bitonic half-cleaner), then 5 bitonic
    // stages sort ascending. All lanes compute (symmetric merge).
    float md[KNBR];
    int   mi[KNBR];
#pragma unroll
    for (int i = 0; i < KNBR; ++i) {
        float b  = __shfl_xor(dist[KNBR - 1 - i], 16, 32);
        int   bi = __shfl_xor(nidx[KNBR - 1 - i], 16, 32);
        bool ta = dist[i] <= b;
        md[i] = ta ? dist[i] : b;
        mi[i] = ta ? nidx[i] : bi;
    }
#pragma unroll
    for (int s = 16; s >= 1; s >>= 1) {
#pragma unroll
        for (int i = 0; i < KNBR; ++i) {
            if ((i & s) == 0) {
                const int j = i + s;
                bool  sw = md[j] < md[i];
                float ta = md[i], tb2 = md[j];
                int   ia = mi[i], ib  = mi[j];
                md[i] = sw ? tb2 : ta;  md[j] = sw ? ta : tb2;
                mi[i] = sw ? ib  : ia;  mi[j] = sw ? ia : ib;
            }
        }
    }

    // ---- emit: lane pair splits the 32 slots; each lane materializes its
    // 16 consecutive edges in registers, then writes aligned B128 stores.
    const int       gctr  = mol * MSIZE + cl;
    const long long ebase = (long long)gctr * KNBR + half * 16;

    int   rowv[16], colv[16];
    float wv[16],  vv[48];
    unsigned int mw[4] = {0u, 0u, 0u, 0u};

#pragma unroll
    for (int j = 0; j < 16; ++j) {
        const int  mloc  = half ? mi[j + 16] : mi[j];   // static-index select
        const bool valid = mloc >= 0;
        const int  msafe = valid ? mloc : 0;
        float4 pn = sP[msafe];
        float vx = valid ? pn.x - pc.x : 0.0f;
        float vy = valid ? pn.y - pc.y : 0.0f;
        float vz = valid ? pn.z - pc.z : 0.0f;
        float d2e = vx * vx + vy * vy + vz * vz;
        float w   = valid ? __builtin_sqrtf(d2e > 0.0f ? d2e : 1.0f) : 0.0f;

        rowv[j] = valid ? (mol * MSIZE + mloc) : -1;
        colv[j] = valid ? gctr : -1;
        wv[j]   = w;
        vv[3 * j + 0] = vx;
        vv[3 * j + 1] = vy;
        vv[3 * j + 2] = vz;
        mw[j >> 2] |= (valid ? 1u : 0u) << ((j & 3) * 8);
    }

    // edge_index rows / cols: 4x int4 each (ebase is a multiple of 16)
    int4* pr = (int4*)(ei_row + ebase);
    int4* pcN = (int4*)(ei_col + ebase);
#pragma unroll
    for (int q = 0; q < 4; ++q) {
        pr[q]  = make_int4(rowv[4*q+0], rowv[4*q+1], rowv[4*q+2], rowv[4*q+3]);
        pcN[q] = make_int4(colv[4*q+0], colv[4*q+1], colv[4*q+2], colv[4*q+3]);
    }
    // edge_weight: 4x float4
    float4* pw = (float4*)(ew + ebase);
#pragma unroll
    for (int q = 0; q < 4; ++q)
        pw[q] = make_float4(wv[4*q+0], wv[4*q+1], wv[4*q+2], wv[4*q+3]);
    // edge_vec: 48 consecutive floats -> 12x float4 (3*ebase is 16B aligned)
    float4* pv = (float4*)(evec + 3 * ebase);
#pragma unroll
    for (int q = 0; q < 12; ++q)
        pv[q] = make_float4(vv[4*q+0], vv[4*q+1], vv[4*q+2], vv[4*q+3]);
    // mask: 16 bytes packed -> one int4
    *(int4*)(emask + ebase) = make_int4((int)mw[0], (int)mw[1], (int)mw[2], (int)mw[3]);
}

extern "C" void kernel_launch(void* const* d_in, const int* in_sizes, int n_in,
                              void* d_out, int out_size, void* d_ws, size_t ws_size,
                              hipStream_t stream)
{
    (void)in_sizes; (void)n_in; (void)d_ws; (void)ws_size; (void)out_size;

    const float* pos = (const float*)d_in[0];   // [N,3] f32; batch (d_in[1]) implicit

    const long long E = (long long)NMOL * MSIZE * KNBR;   // 4,194,304
    int*           ei_row = (int*)d_out;                  // edge_index[0]
    int*           ei_col = ei_row + E;                   // edge_index[1]
    float*         ew     = (float*)(ei_col + E);         // edge_weight
    float*         evec   = ew + E;                       // edge_vec [E,3]
    unsigned char* emask  = (unsigned char*)(evec + 3 * E);

    dim3 grid(NMOL);
    dim3 block(1024);   // 32 wave32s: wave w handles centers [16w, 16w+16)
    hipLaunchKernelGGL(radius_knn_kernel, grid, block, 0, stream,
                       pos, ei_row, ei_col, ew, evec, emask);
}